// deeds_3934190044187
// MI455X (gfx1250) — compile-verified
//
#include <hip/hip_runtime.h>
#include <hip/hip_bf16.h>

// deeds two-stage cost-volume smoothing, fp32, for MI455X (gfx1250).
//   label smoothing (per voxel, 7^3 cube): reppad3 -> max3 -> avg3 -> avg3
//   spatial smoothing (per label, 40^3):   reppad2 -> avg3 -> avg3
// avg3∘avg3 == separable 5-tap triangle [1,2,3,2,1]/9 per axis; the clamped
// spatial triangle is a banded 40x40 matrix => V_WMMA_F32_16X16X4_F32.
// Memory/L2-bandwidth bound (~1.4 GB traffic, ping-pong fits in 192MB L2).

typedef __attribute__((ext_vector_type(2))) float v2f;
typedef __attribute__((ext_vector_type(8))) float v8f;

#define XG 40
#define LQ 343            // 7^3
#define NVOX 64000        // 40^3
#define VPB 4             // voxels per block in label kernel

static constexpr float TRI0 = 1.f / 9.f, TRI1 = 2.f / 9.f, TRI2 = 3.f / 9.f;
static constexpr float TRIC[5] = {TRI0, TRI1, TRI2, TRI1, TRI0};

// Clamped 5-tap triangle band matrix W[i][j] along a length-40 axis,
// padded to 48 rows (rows 40..47 zero) for the mt=2 WMMA tile.
struct WTab { float w[48][40]; };
static constexpr WTab make_wtab() {
    WTab t{};
    for (int i = 0; i < 48; ++i)
        for (int j = 0; j < 40; ++j) {
            float w = 0.f;
            if (i < 40)
                for (int d = 0; d < 5; ++d) {
                    int jj = i + d - 2;
                    jj = jj < 0 ? 0 : (jj > 39 ? 39 : jj);
                    if (jj == j) w += TRIC[d];
                }
            t.w[i][j] = w;
        }
    return t;
}
__constant__ WTab WT = make_wtab();

// ---------------------------------------------------------------------------
// Label-space smoothing: one block = 4 voxels, staged through LDS.
// stage==0: in = a1 + a0*cost
// stage==1: in = a4 + a2*(a1 + a0*cost) + a3*avgIn
// Separable: clamped max along a,b,c -> 11^3, then VALID triangle -> 7^3.
// ---------------------------------------------------------------------------
__global__ void __launch_bounds__(128)
label_smooth_kernel(const float* __restrict__ cost,
                    const float* __restrict__ avgIn,
                    float* __restrict__ out,
                    const float* __restrict__ alpha,
                    int stage)
{
    __shared__ float raw[VPB][344];
    __shared__ float t1[VPB][1332];
    __shared__ float t2[VPB][1332];

    const int tid = threadIdx.x;
    const int vb  = blockIdx.x * VPB;  // 16000 blocks * 4 == 64000 exactly

    const float a0 = alpha[0], a1 = alpha[1];
    const float a2 = alpha[2], a3 = alpha[3], a4 = alpha[4];

    // P0: load + affine
    for (int s = 0; s < VPB; ++s)
        for (int l = tid; l < LQ; l += 128) {
            int idx = (vb + s) * LQ + l;
            float d = a1 + a0 * cost[idx];
            raw[s][l] = (stage == 0) ? d : (a4 + a2 * d + a3 * avgIn[idx]);
        }
    __syncthreads();

    // P1: clamped max along a: t1[(ia*7+b)*7+c], ia 0..10 -> 539/voxel
    for (int s = 0; s < VPB; ++s)
        for (int r = tid; r < 539; r += 128) {
            int ia = r / 49, bc = r % 49;
            int lo = ia - 3; lo = lo < 0 ? 0 : (lo > 6 ? 6 : lo);
            int hi = ia - 1; hi = hi < 0 ? 0 : (hi > 6 ? 6 : hi);
            float m = raw[s][lo * 49 + bc];
            for (int j = lo + 1; j <= hi; ++j) m = fmaxf(m, raw[s][j * 49 + bc]);
            t1[s][r] = m;
        }
    __syncthreads();

    // P2: clamped max along b: t2[(ia*11+ib)*7+c] -> 847/voxel
    for (int s = 0; s < VPB; ++s)
        for (int r = tid; r < 847; r += 128) {
            int c = r % 7, ib = (r / 7) % 11, ia = r / 77;
            int lo = ib - 3; lo = lo < 0 ? 0 : (lo > 6 ? 6 : lo);
            int hi = ib - 1; hi = hi < 0 ? 0 : (hi > 6 ? 6 : hi);
            float m = t1[s][ia * 49 + lo * 7 + c];
            for (int j = lo + 1; j <= hi; ++j) m = fmaxf(m, t1[s][ia * 49 + j * 7 + c]);
            t2[s][r] = m;
        }
    __syncthreads();

    // P3: clamped max along c: t1[(ia*11+ib)*11+ic] -> 1331/voxel
    for (int s = 0; s < VPB; ++s)
        for (int r = tid; r < 1331; r += 128) {
            int ic = r % 11, ib = (r / 11) % 11, ia = r / 121;
            int lo = ic - 3; lo = lo < 0 ? 0 : (lo > 6 ? 6 : lo);
            int hi = ic - 1; hi = hi < 0 ? 0 : (hi > 6 ? 6 : hi);
            float m = t2[s][(ia * 11 + ib) * 7 + lo];
            for (int j = lo + 1; j <= hi; ++j) m = fmaxf(m, t2[s][(ia * 11 + ib) * 7 + j]);
            t1[s][r] = m;
        }
    __syncthreads();

    // P4: VALID triangle along a: t2[(ja*11+ib)*11+ic], ja 0..6 -> 847/voxel
    for (int s = 0; s < VPB; ++s)
        for (int r = tid; r < 847; r += 128) {
            int ic = r % 11, ib = (r / 11) % 11, ja = r / 121;
            float acc = 0.0f;
            #pragma unroll
            for (int d = 0; d < 5; ++d)
                acc += TRIC[d] * t1[s][((ja + d) * 11 + ib) * 11 + ic];
            t2[s][r] = acc;
        }
    __syncthreads();

    // P5: triangle along b: t1[(ja*7+jb)*11+ic] -> 539/voxel
    for (int s = 0; s < VPB; ++s)
        for (int r = tid; r < 539; r += 128) {
            int ic = r % 11, jb = (r / 11) % 7, ja = r / 77;
            float acc = 0.0f;
            #pragma unroll
            for (int d = 0; d < 5; ++d)
                acc += TRIC[d] * t2[s][(ja * 11 + jb + d) * 11 + ic];
            t1[s][r] = acc;
        }
    __syncthreads();

    // P6: triangle along c + store
    for (int s = 0; s < VPB; ++s)
        for (int l = tid; l < LQ; l += 128) {
            int jc = l % 7, jb = (l / 7) % 7, ja = l / 49;
            float acc = 0.0f;
            #pragma unroll
            for (int d = 0; d < 5; ++d)
                acc += TRIC[d] * t1[s][(ja * 7 + jb) * 11 + jc + d];
            out[(vb + s) * LQ + l] = acc;
        }
}

// ---------------------------------------------------------------------------
// Spatial smoothing axis pass as banded 40x40 matmul per line:
//   out(i, label) = sum_j W[i][j] * in(j, label)
// One wave handles one (line q, 16-label tile): all 3 row tiles at once, so
// each of the 10 K-chunks is loaded exactly once; band predicates are
// compile-time under full unroll => 14 static V_WMMA_F32_16X16X4_F32.
//   mt=0 rows 0..15  needs chunks 0..4
//   mt=1 rows 16..31 needs chunks 3..8
//   mt=2 rows 32..39 needs chunks 7..9
// mode 0: z (stride 1 voxel), 1: y (stride 40), 2: x (stride 1600).
// Waves: 1600 lines * 22 label tiles = 35200 -> 4400 blocks * 8 waves.
// ---------------------------------------------------------------------------
__global__ void __launch_bounds__(256)
spatial_pass_kernel(const float* __restrict__ in,
                    float* __restrict__ out,
                    const float* __restrict__ alpha,
                    int mode, int apply_scale)
{
    const int lane = threadIdx.x & 31;
    const int wave = threadIdx.x >> 5;
    const int tile = blockIdx.x * 8 + wave;      // < 35200 exactly

    const int nt = tile % 22;
    const int q  = tile / 22;                    // line id, < 1600

    int baseVox, sVox;
    if (mode == 0)      { baseVox = q * XG;                      sVox = 1;    }
    else if (mode == 1) { baseVox = (q / XG) * 1600 + (q % XG);  sVox = XG;   }
    else                { baseVox = q;                           sVox = 1600; }

    const int base = baseVox * LQ;
    const int S    = sVox * LQ;                  // element stride along axis

    const int half = lane >> 4;                  // 0: lanes 0-15, 1: 16-31
    const int mrow = lane & 15;
    const int col  = nt * 16 + (lane & 15);      // label column
    // Clamp the load column instead of zeroing: garbage only reaches output
    // columns >= 343, which are never stored. Keeps loads divergence-free.
    const int colc = col < LQ ? col : (LQ - 1);
    const bool colOK = (col < LQ);

    v8f c0 = {}, c1 = {}, c2 = {};
    #pragma unroll
    for (int kc = 0; kc < 10; ++kc) {
        const int k0 = kc * 4 + half * 2;        // this lane's two K slots
        v2f b;
        // B 4x16 f32: VGPR r holds rows r (lanes 0-15) and r+2 (lanes 16-31)
        b.x = in[base + k0 * S + colc];
        b.y = in[base + (k0 + 1) * S + colc];
        // A 16x4 f32: lanes 0-15 hold {K0,K1}, lanes 16-31 hold {K2,K3}
        if (kc <= 4) {                            // row tile 0 (rows 0..15)
            v2f a; a.x = WT.w[mrow][k0];      a.y = WT.w[mrow][k0 + 1];
            c0 = __builtin_amdgcn_wmma_f32_16x16x4_f32(false, a, false, b,
                                                       (short)0, c0, false, false);
        }
        if (kc >= 3 && kc <= 8) {                 // row tile 1 (rows 16..31)
            v2f a; a.x = WT.w[16 + mrow][k0]; a.y = WT.w[16 + mrow][k0 + 1];
            c1 = __builtin_amdgcn_wmma_f32_16x16x4_f32(false, a, false, b,
                                                       (short)0, c1, false, false);
        }
        if (kc >= 7) {                            // row tile 2 (rows 32..39)
            v2f a; a.x = WT.w[32 + mrow][k0]; a.y = WT.w[32 + mrow][k0 + 1];
            c2 = __builtin_amdgcn_wmma_f32_16x16x4_f32(false, a, false, b,
                                                       (short)0, c2, false, false);
        }
    }

    const float scale = apply_scale ? alpha[5] : 1.0f;
    // C/D 16x16 f32: VGPR r holds row r (lanes 0-15) / row r+8 (lanes 16-31)
    #pragma unroll
    for (int r = 0; r < 8; ++r) {
        int row = r + half * 8;
        if (colOK) {
            out[base + row * S + col]        = scale * c0[r];
            out[base + (16 + row) * S + col] = scale * c1[r];
            if (row + 32 < XG)
                out[base + (32 + row) * S + col] = scale * c2[r];
        }
    }
}

// ---------------------------------------------------------------------------
extern "C" void kernel_launch(void* const* d_in, const int* in_sizes, int n_in,
                              void* d_out, int out_size, void* d_ws, size_t ws_size,
                              hipStream_t stream) {
    const float* cost  = (const float*)d_in[0];   // (64000, 343) f32
    const float* alpha = (const float*)d_in[1];   // (6,) f32
    float* out = (float*)d_out;                   // (64000*343) f32

    const long N = (long)NVOX * LQ;
    float* bufA = (float*)d_ws;
    float* bufB = bufA + N;                       // needs 2*N*4 = ~176 MB of d_ws

    const dim3 gL(NVOX / VPB), bL(128);           // label smoothing
    const dim3 gS(4400), bS(256);                 // spatial WMMA passes

    // stage 1
    label_smooth_kernel<<<gL, bL, 0, stream>>>(cost, nullptr, bufA, alpha, 0);
    spatial_pass_kernel<<<gS, bS, 0, stream>>>(bufA, bufB, alpha, 0, 0); // z
    spatial_pass_kernel<<<gS, bS, 0, stream>>>(bufB, bufA, alpha, 1, 0); // y
    spatial_pass_kernel<<<gS, bS, 0, stream>>>(bufA, bufB, alpha, 2, 0); // x
    // stage 2 (deeds recomputed from cost inside the kernel)
    label_smooth_kernel<<<gL, bL, 0, stream>>>(cost, bufB, bufA, alpha, 1);
    spatial_pass_kernel<<<gS, bS, 0, stream>>>(bufA, bufB, alpha, 0, 0); // z
    spatial_pass_kernel<<<gS, bS, 0, stream>>>(bufB, bufA, alpha, 1, 0); // y
    spatial_pass_kernel<<<gS, bS, 0, stream>>>(bufA, out,  alpha, 2, 1); // x + alpha5
}